// GatedGraphConvolution_1726576856964
// MI455X (gfx1250) — compile-verified
//
#include <hip/hip_runtime.h>

// Gated Graph Convolution, restructured:
//   Y = X @ W.T          (per-node, 50000x128 @ 128x256 -> WMMA f32 GEMM, 3.3 GFLOP)
//   per edge e: msg = sigmoid(Y[tgt,0:128]) * Y[tgt,128:256]
//               out[src] += msg   (hardware f32 atomics through L2)
//   out initialized to input.
// Everything (X 25.6MB, Y 51.2MB, W 128KB, out 25.6MB) is L2-resident on MI455X
// (192MB L2), so the per-edge random gathers/atomics stay on-chip.

typedef __attribute__((ext_vector_type(2))) float v2f;
typedef __attribute__((ext_vector_type(8))) float v8f;

#define NNODES 50000
#define NEDGES 800000
#define INF    128
#define OUTF   128
#define TWOOUT 256
// 50000 = 3125 * 16 exactly: no M-edge padding needed.
#define MTILES 3125

// ---------------------------------------------------------------------------
// Kernel 1: out = input  (float4 copy)
// ---------------------------------------------------------------------------
__global__ __launch_bounds__(256) void init_out_kernel(const float* __restrict__ in,
                                                       float* __restrict__ out,
                                                       int n4) {
  int i = blockIdx.x * blockDim.x + threadIdx.x;
  if (i < n4) {
    ((float4*)out)[i] = ((const float4*)in)[i];
  }
}

// ---------------------------------------------------------------------------
// Kernel 2: Y[n, o] = sum_k X[n,k] * W[o,k]   (Y = X @ W^T), f32 WMMA.
//
// One wave32 computes a 16(M) x 64(N) strip: the A fragment is loaded ONCE
// per K-step and reused across 4 V_WMMA_F32_16X16X4_F32 into 4 independent
// accumulators -> 4 WMMAs per 5 float2 loads (vs 1 per 2 before), 4x less
// A-operand L2 traffic. Block = 4 waves = one M-tile x all 256 N columns.
//
// A-frag (16x4, MxK): lane L holds row M = L&15; K pair {0,1} for lanes 0-15,
//   {2,3} for lanes 16-31 -> one float2 load per step.
// B-frag (4x16, KxN): same K split, N = lane&15; B[k][n] = W[n*128 + k].
// D (16x16 f32): VGPR r -> row m0 + r + (lane>=16 ? 8 : 0), col n0 + (lane&15).
// ---------------------------------------------------------------------------
__global__ __launch_bounds__(128) void node_gemm_wmma_kernel(const float* __restrict__ X,
                                                             const float* __restrict__ W,
                                                             float* __restrict__ Y) {
  const int wave = threadIdx.x >> 5;       // 4 waves: N-groups 0..3 (64 cols each)
  const int lane = threadIdx.x & 31;
  const int m0 = blockIdx.x * 16;          // one M-tile per block
  const int n0 = wave * 64;                // this wave's 64-column strip

  const int r     = lane & 15;
  const int khalf = (lane >> 4) << 1;      // 0 for lanes 0-15, 2 for 16-31

  const float* arow = X + (size_t)(m0 + r) * INF + khalf;
  const float* b0   = W + (size_t)(n0 +  0 + r) * INF + khalf;
  const float* b1   = W + (size_t)(n0 + 16 + r) * INF + khalf;
  const float* b2   = W + (size_t)(n0 + 32 + r) * INF + khalf;
  const float* b3   = W + (size_t)(n0 + 48 + r) * INF + khalf;

  v8f acc0 = {}, acc1 = {}, acc2 = {}, acc3 = {};
#pragma unroll 4
  for (int k = 0; k < INF; k += 4) {
    v2f a  = *(const v2f*)(arow + k);
    v2f w0 = *(const v2f*)(b0 + k);
    v2f w1 = *(const v2f*)(b1 + k);
    v2f w2 = *(const v2f*)(b2 + k);
    v2f w3 = *(const v2f*)(b3 + k);
    // 8 args: (neg_a, A, neg_b, B, c_mod, C, reuse_a, reuse_b)
    acc0 = __builtin_amdgcn_wmma_f32_16x16x4_f32(false, a, false, w0, (short)0, acc0, false, false);
    acc1 = __builtin_amdgcn_wmma_f32_16x16x4_f32(false, a, false, w1, (short)0, acc1, false, false);
    acc2 = __builtin_amdgcn_wmma_f32_16x16x4_f32(false, a, false, w2, (short)0, acc2, false, false);
    acc3 = __builtin_amdgcn_wmma_f32_16x16x4_f32(false, a, false, w3, (short)0, acc3, false, false);
  }

  const int rowOff = (lane >> 4) * 8;      // high half-wave -> M += 8
  float* ybase = Y + (size_t)(m0 + rowOff) * TWOOUT + n0 + r;
#pragma unroll
  for (int i = 0; i < 8; ++i) {
    ybase[(size_t)i * TWOOUT +  0] = acc0[i];
    ybase[(size_t)i * TWOOUT + 16] = acc1[i];
    ybase[(size_t)i * TWOOUT + 32] = acc2[i];
    ybase[(size_t)i * TWOOUT + 48] = acc3[i];
  }
}

// ---------------------------------------------------------------------------
// Kernel 3: per-edge gather + gated scatter-add.
// One wave32 per edge; each lane owns 4 features (32*4 = 128).
// Reads Y[tgt] as float4s (L2-resident), 4 hardware f32 atomics per lane.
// ---------------------------------------------------------------------------
__device__ __forceinline__ float sigmoidf(float x) {
  return 1.0f / (1.0f + __expf(-x));
}

__global__ __launch_bounds__(256) void edge_scatter_kernel(const float* __restrict__ Y,
                                                           const int* __restrict__ esrc,
                                                           const int* __restrict__ etgt,
                                                           float* __restrict__ out) {
  const int e = blockIdx.x * 8 + (threadIdx.x >> 5);
  const int lane = threadIdx.x & 31;
  if (e >= NEDGES) return;

  const int tgt = etgt[e];
  const int src = esrc[e];

  const float4* yg = (const float4*)(Y + (size_t)tgt * TWOOUT);
  float4 g  = yg[lane];        // gate part: features [lane*4, lane*4+4)
  float4 ev = yg[32 + lane];   // edge part: offset by 128 floats = 32 float4

  float4 m;
  m.x = ev.x * sigmoidf(g.x);
  m.y = ev.y * sigmoidf(g.y);
  m.z = ev.z * sigmoidf(g.z);
  m.w = ev.w * sigmoidf(g.w);

  float* op = out + (size_t)src * OUTF + lane * 4;
  unsafeAtomicAdd(op + 0, m.x);   // -> global_atomic_add_f32 (no return)
  unsafeAtomicAdd(op + 1, m.y);
  unsafeAtomicAdd(op + 2, m.z);
  unsafeAtomicAdd(op + 3, m.w);
}

// ---------------------------------------------------------------------------
extern "C" void kernel_launch(void* const* d_in, const int* in_sizes, int n_in,
                              void* d_out, int out_size, void* d_ws, size_t ws_size,
                              hipStream_t stream) {
  const float* X    = (const float*)d_in[0];   // [50000, 128]
  const float* W    = (const float*)d_in[1];   // [256, 128]
  const int*   esrc = (const int*)d_in[2];     // [800000]
  const int*   etgt = (const int*)d_in[3];     // [800000]
  float*       out  = (float*)d_out;           // [50000, 128]
  float*       Y    = (float*)d_ws;            // [50000, 256] = 51.2 MB scratch

  // 1) out = input
  const int n4 = NNODES * INF / 4;
  init_out_kernel<<<(n4 + 255) / 256, 256, 0, stream>>>(X, out, n4);

  // 2) Y = X @ W^T via f32 WMMA. One block (4 waves) per M-tile; each wave
  //    computes 16x64 with a shared A fragment across 4 accumulators.
  node_gemm_wmma_kernel<<<MTILES, 128, 0, stream>>>(X, W, Y);

  // 3) gather + gate + atomic scatter-add. One wave per edge, 8 edges/block.
  edge_scatter_kernel<<<(NEDGES + 7) / 8, 256, 0, stream>>>(Y, esrc, etgt, out);
}